// Lineq2v2_35725537968676
// MI455X (gfx1250) — compile-verified
//
#include <hip/hip_runtime.h>
#include <cstdint>

#define B_ 32
#define N_ 128
#define L_ 32

typedef float v2f __attribute__((ext_vector_type(2)));
typedef float v4f __attribute__((ext_vector_type(4)));
typedef float v8f __attribute__((ext_vector_type(8)));

// ---- CDNA5 async global->LDS copy (ASYNCcnt-tracked), per-lane 16B ----
__device__ __forceinline__ void async_ld_b128(uint32_t lds_byte_off, const void* gaddr) {
    asm volatile("global_load_async_to_lds_b128 %0, %1, off"
                 :: "v"(lds_byte_off), "v"(gaddr)
                 : "memory");
}
__device__ __forceinline__ void wait_async0() {
    asm volatile("s_wait_asynccnt 0" ::: "memory");
}
// Low 32 bits of a generic pointer to __shared__ == LDS byte offset.
__device__ __forceinline__ uint32_t lds_off(const void* p) {
    return (uint32_t)(uintptr_t)p;
}
__device__ __forceinline__ v4f ld4(const float* p) {
    return *(const v4f*)p;
}

// ---- WMMA column-sum: dst[n] = sum_{k=0..127} src[k*stride + n], n = lane&15 ----
// Uses D = ones(16x4) x B(4x16) + C with V_WMMA_F32_16X16X4_F32.
// All-ones A makes the result invariant to the B fragment's K<->lane mapping,
// and every row of D is identical, so c[0] on lanes 0..15 is the answer.
__device__ __forceinline__ void wmma_reduce_128(const float* __restrict__ src,
                                                size_t stride,
                                                float* __restrict__ dst,
                                                int lane) {
    const int sub = lane >> 4;      // which K pair this half-wave supplies
    const int n   = lane & 15;      // column (= l offset)
    const v2f ones = {1.f, 1.f};
    v8f c = {0.f, 0.f, 0.f, 0.f, 0.f, 0.f, 0.f, 0.f};
#pragma unroll 4
    for (int k0 = 0; k0 < N_; k0 += 4) {
        v2f bfrag;
        bfrag.x = src[(size_t)(k0 + sub)     * stride + n];
        bfrag.y = src[(size_t)(k0 + 2 + sub) * stride + n];
        c = __builtin_amdgcn_wmma_f32_16x16x4_f32(
                /*neg_a=*/false, ones, /*neg_b=*/false, bfrag,
                /*c_mod=*/(short)0, c, /*reuse_a=*/false, /*reuse_b=*/false);
    }
    if (lane < 16) dst[lane] = c[0];
}

// ---- Kernel 1: rowsum / colsum (WMMA) + diag, one block per (b, r) ----
__global__ __launch_bounds__(128)
void reduce_rows_cols(const float* __restrict__ in,
                      float* __restrict__ rs, float* __restrict__ cs,
                      float* __restrict__ dgv) {
    const int b = blockIdx.x >> 7;
    const int r = blockIdx.x & (N_ - 1);
    const int tid  = threadIdx.x;
    const int lane = tid & 31;
    const int wv   = __builtin_amdgcn_readfirstlane(tid >> 5);  // scalar branch
    const int l0   = (wv & 1) << 4;                             // l-half: 0 or 16

    const float* inB = in + (size_t)b * (N_ * N_ * L_);
    if (wv < 2) {   // rowsum[b,r,l] = sum_j in[b,r,j,l]
        wmma_reduce_128(inB + (size_t)r * (N_ * L_) + l0, L_,
                        rs + (size_t)blockIdx.x * L_ + l0, lane);
    } else {        // colsum[b,r,l] = sum_i in[b,i,r,l]
        wmma_reduce_128(inB + (size_t)r * L_ + l0, (size_t)N_ * L_,
                        cs + (size_t)blockIdx.x * L_ + l0, lane);
    }
    if (tid < L_)   // diag[b,r,l] = in[b,r,r,l]
        dgv[(size_t)blockIdx.x * L_ + tid] =
            inB[(size_t)r * (N_ * L_) + (size_t)r * L_ + tid];
}

// ---- Kernel 2: totsum / trace (WMMA), one block per b ----
__global__ __launch_bounds__(128)
void reduce_scalars(const float* __restrict__ rs, const float* __restrict__ dgv,
                    float* __restrict__ tot, float* __restrict__ trc) {
    const int b    = blockIdx.x;
    const int tid  = threadIdx.x;
    const int lane = tid & 31;
    const int wv   = __builtin_amdgcn_readfirstlane(tid >> 5);
    const int l0   = (wv & 1) << 4;

    if (wv < 2)     // totsum[b,l] = sum_i rowsum[b,i,l]
        wmma_reduce_128(rs + (size_t)b * (N_ * L_) + l0, L_,
                        tot + (size_t)b * L_ + l0, lane);
    else            // trace[b,l] = sum_i diag[b,i,l]
        wmma_reduce_128(dgv + (size_t)b * (N_ * L_) + l0, L_,
                        trc + (size_t)b * L_ + l0, lane);
}

// ---- Kernel 3: build all 15 segments; one block per (b, i) ----
__global__ __launch_bounds__(512)
void writer(const float* __restrict__ in,
            const float* __restrict__ rs, const float* __restrict__ cs,
            const float* __restrict__ dgv,
            const float* __restrict__ tot, const float* __restrict__ trc,
            float* __restrict__ out) {
    __shared__ __align__(16) float SR[N_ * L_];   // rowsum[b, j, l]
    __shared__ __align__(16) float SC[N_ * L_];   // colsum[b, j, l]
    __shared__ __align__(16) float SD[N_ * L_];   // diag  [b, j, l]
    __shared__ __align__(16) float ST[N_ * L_];   // in[b, j, i, l]  (transposed column)
    __shared__ __align__(16) float Stot[L_];
    __shared__ __align__(16) float Strc[L_];

    const int b   = blockIdx.x >> 7;
    const int i   = blockIdx.x & (N_ - 1);
    const int tid = threadIdx.x;

    const float* inB = in  + (size_t)b * (N_ * N_ * L_);
    const float* rsb = rs  + (size_t)b * (N_ * L_);
    const float* csb = cs  + (size_t)b * (N_ * L_);
    const float* dgb = dgv + (size_t)b * (N_ * L_);

    const uint32_t sr0 = lds_off(SR), sc0 = lds_off(SC);
    const uint32_t sd0 = lds_off(SD), st0 = lds_off(ST);

    // 16 KB per array = 1024 chunks of 16B; 512 threads -> 2 chunks each.
    for (int t = tid; t < (N_ * L_ * 4) / 16; t += 512) {
        const uint32_t bo = (uint32_t)t * 16u;
        async_ld_b128(sr0 + bo, (const char*)rsb + bo);
        async_ld_b128(sc0 + bo, (const char*)csb + bo);
        async_ld_b128(sd0 + bo, (const char*)dgb + bo);
        const int j = t >> 3, q = t & 7;   // strided transpose column gather
        async_ld_b128(st0 + bo,
                      (const char*)(inB + ((size_t)j * N_ + i) * L_) + q * 16);
    }
    if (tid < L_) {
        Stot[tid] = tot[b * L_ + tid];
        Strc[tid] = trc[b * L_ + tid];
    }
    wait_async0();
    __syncthreads();

    const float* inRow  = inB + (size_t)i * (N_ * L_);          // in[b,i,j,l]
    float*       outRow = out + (size_t)blockIdx.x * N_ * 480;  // out[b,i,*,*]
    const v4f z4 = (v4f){0.f, 0.f, 0.f, 0.f};

    // 128 j * 15 seg * 8 quads = 15360 float4 NT stores per block.
    for (int f = tid; f < N_ * 120; f += 512) {
        const int j   = f / 120;
        const int r   = f - j * 120;
        const int seg = r >> 3;
        const int l0  = (r & 7) << 2;
        const bool dg = (i == j);
        v4f v;
        switch (seg) {
            case 0:  v = dg ? ld4(&Stot[l0]) : z4;            break;  // totsum*eye
            case 1:  v = ld4(&Stot[l0]);                      break;  // totsum
            case 2:  v = dg ? ld4(&Strc[l0]) : z4;            break;  // trace*eye
            case 3:  v = ld4(&Strc[l0]);                      break;  // trace
            case 4:  v = ld4(&SR[i * L_ + l0]);               break;  // rowsum[b,i,l]
            case 5:  v = ld4(&SR[j * L_ + l0]);               break;  // rowsum[b,j,l]
            case 6:  v = dg ? ld4(&SR[i * L_ + l0]) : z4;     break;  // rowsum*eye
            case 7:  v = ld4(&SC[i * L_ + l0]);               break;  // colsum[b,i,l]
            case 8:  v = ld4(&SC[j * L_ + l0]);               break;  // colsum[b,j,l]
            case 9:  v = dg ? ld4(&SC[i * L_ + l0]) : z4;     break;  // colsum*eye
            case 10: v = ld4(&SD[i * L_ + l0]);               break;  // diag[b,i,l]
            case 11: v = ld4(&SD[j * L_ + l0]);               break;  // diag[b,j,l]
            case 12: v = dg ? ld4(&SD[i * L_ + l0]) : z4;     break;  // diag*eye
            case 13: v = *(const v4f*)(inRow + (size_t)j * L_ + l0); break; // inputs
            default: v = ld4(&ST[j * L_ + l0]);               break;  // inputs^T
        }
        __builtin_nontemporal_store(
            v, (v4f*)(outRow + (size_t)j * 480 + (seg << 5) + l0));
    }
}

extern "C" void kernel_launch(void* const* d_in, const int* in_sizes, int n_in,
                              void* d_out, int out_size, void* d_ws, size_t ws_size,
                              hipStream_t stream) {
    (void)in_sizes; (void)n_in; (void)out_size; (void)ws_size;
    const float* in = (const float*)d_in[0];
    float* out = (float*)d_out;

    float* rs  = (float*)d_ws;                 // B*N*L = 131072 floats
    float* cs  = rs  + (size_t)B_ * N_ * L_;   // 131072
    float* dgv = cs  + (size_t)B_ * N_ * L_;   // 131072
    float* tot = dgv + (size_t)B_ * N_ * L_;   // 1024
    float* trc = tot + (size_t)B_ * L_;        // 1024

    reduce_rows_cols<<<B_ * N_, 128, 0, stream>>>(in, rs, cs, dgv);
    reduce_scalars<<<B_, 128, 0, stream>>>(rs, dgv, tot, trc);
    writer<<<B_ * N_, 512, 0, stream>>>(in, rs, cs, dgv, tot, trc, out);
}